// KVCacheKernel_88562225643902
// MI455X (gfx1250) — compile-verified
//
#include <hip/hip_runtime.h>

// KV-cache scatter+gather for MI455X (gfx1250).
// Pure bandwidth kernel: 128MB total traffic @ 23.3 TB/s => ~5.5us floor.
// Zero FLOPs => WMMA not applicable; the CDNA5 paths that matter are b128
// coalesced vmem and the async global<->LDS DMA ops (ASYNCcnt).

typedef unsigned int u32x4 __attribute__((ext_vector_type(4)));
typedef int          i32x4 __attribute__((ext_vector_type(4)));

#define AS1 __attribute__((address_space(1)))
#define AS3 __attribute__((address_space(3)))

#ifndef __has_builtin
#define __has_builtin(x) 0
#endif

#if __has_builtin(__builtin_amdgcn_global_load_async_to_lds_b128) && \
    __has_builtin(__builtin_amdgcn_global_store_async_from_lds_b128)
#define KV_ASYNC_LDS 1
#else
#define KV_ASYNC_LDS 0
#endif

__device__ __forceinline__ void kv_wait_asynccnt0() {
#if __has_builtin(__builtin_amdgcn_s_wait_asynccnt)
  __builtin_amdgcn_s_wait_asynccnt(0);
#else
  asm volatile("s_wait_asynccnt 0" ::: "memory");
#endif
}

// Problem constants (fixed by the reference).
static constexpr int kH = 32;        // heads
static constexpr int kD = 128;       // head dim (halves)
static constexpr int kSmax = 4096;   // cache max seq len
static constexpr int kChunksPerTok = kH * kD / 8;  // 512 (16B chunks per token)
static constexpr int kChunksPerRow = kD / 8;       // 16  (16B chunks per head row)

// --- tiny prep kernels: build inverse position map in workspace ------------
__global__ void kv_inv_init(int* __restrict__ inv, int smax) {
  int i = blockIdx.x * blockDim.x + threadIdx.x;
  if (i < smax) inv[i] = -1;
}

__global__ void kv_inv_scatter(const int* __restrict__ pos,
                               int* __restrict__ inv, int s) {
  int i = blockIdx.x * blockDim.x + threadIdx.x;
  if (i < s) {
    int p = pos[i];
    if (p >= 0 && p < kSmax) inv[p] = i;
  }
}

// --- main streaming gather -------------------------------------------------
// out layout: [K: S*H*D halves][V: S*H*D halves]; both [S,H,D] row-major.
// Each thread moves one 16B chunk of K and the matching chunk of V.
__global__ __launch_bounds__(256) void kv_gather(
    const u32x4* __restrict__ keys, const u32x4* __restrict__ vals,
    const u32x4* __restrict__ kcache, const u32x4* __restrict__ vcache,
    const int* __restrict__ inv, const int* __restrict__ bidx,
    u32x4* __restrict__ out, int nchunks) {
  int n = blockIdx.x * 256 + threadIdx.x;
  if (n >= nchunks) return;
  int s   = n >> 9;       // n / kChunksPerTok
  int rem = n & 511;      // chunk within token: h*16 + c
  int h   = rem >> 4;
  int c   = rem & 15;

  int j = inv[s];
  const u32x4* sk;
  const u32x4* sv;
  if (j >= 0) {
    // Position was freshly scattered: source is keys/values (same layout as out).
    long long o = (long long)j * kChunksPerTok + rem;
    sk = keys + o;
    sv = vals + o;
  } else {
    // Untouched position: source is the original cache row for batch_idx.
    int b = *bidx;  // device scalar; cold path only
    long long o = ((long long)(b * kH + h) * kSmax + s) * kChunksPerRow + c;
    sk = kcache + o;
    sv = vcache + o;
  }

#if KV_ASYNC_LDS
  // CDNA5 async DMA path: global -> LDS -> global, tracked by ASYNCcnt.
  __shared__ u32x4 bk[256];
  __shared__ u32x4 bv[256];
  AS3 i32x4* lbk = (AS3 i32x4*)(void*)&bk[threadIdx.x];
  AS3 i32x4* lbv = (AS3 i32x4*)(void*)&bv[threadIdx.x];
  __builtin_amdgcn_global_load_async_to_lds_b128((AS1 i32x4*)sk, lbk, 0, 0);
  __builtin_amdgcn_global_load_async_to_lds_b128((AS1 i32x4*)sv, lbv, 0, 0);
  kv_wait_asynccnt0();  // LDS filled before draining
  __builtin_amdgcn_global_store_async_from_lds_b128((AS1 i32x4*)(out + n), lbk, 0, 0);
  __builtin_amdgcn_global_store_async_from_lds_b128((AS1 i32x4*)(out + nchunks + n), lbv, 0, 0);
  // outstanding async stores are waited implicitly at S_ENDPGM
#else
  // Fallback: non-temporal b128 stream (data touched exactly once).
  u32x4 a  = __builtin_nontemporal_load(sk);
  u32x4 b2 = __builtin_nontemporal_load(sv);
  __builtin_nontemporal_store(a,  out + n);
  __builtin_nontemporal_store(b2, out + nchunks + n);
#endif
}

extern "C" void kernel_launch(void* const* d_in, const int* in_sizes, int n_in,
                              void* d_out, int out_size, void* d_ws, size_t ws_size,
                              hipStream_t stream) {
  const u32x4* keys   = (const u32x4*)d_in[0];
  const u32x4* vals   = (const u32x4*)d_in[1];
  const u32x4* kcache = (const u32x4*)d_in[2];
  const u32x4* vcache = (const u32x4*)d_in[3];
  const int*   pos    = (const int*)d_in[4];
  const int*   bidx   = (const int*)d_in[5];  // 1-element device array

  int S       = in_sizes[4];        // 4096 tokens
  int nchunks = in_sizes[0] / 8;    // S*H*D/8 16B chunks = 2,097,152

  int* inv = (int*)d_ws;            // kSmax ints = 16 KB of workspace

  kv_inv_init<<<(kSmax + 255) / 256, 256, 0, stream>>>(inv, kSmax);
  kv_inv_scatter<<<(S + 255) / 256, 256, 0, stream>>>(pos, inv, S);
  kv_gather<<<(nchunks + 255) / 256, 256, 0, stream>>>(
      keys, vals, kcache, vcache, inv, bidx, (u32x4*)d_out, nchunks);
}